// GCNModule_59390807769620
// MI455X (gfx1250) — compile-verified
//
#include <hip/hip_runtime.h>

#define NN 50000
#define NE 800000
#define DD 128

typedef float v2f __attribute__((ext_vector_type(2)));
typedef float v8f __attribute__((ext_vector_type(8)));

// ---------------- utility: zero a float buffer ----------------
__global__ void zero_f32(float* __restrict__ p, int n) {
  int i = blockIdx.x * blockDim.x + threadIdx.x;
  if (i < n) p[i] = 0.0f;
}

// ---------------- degrees: deg[src[e]] += 1 ----------------
__global__ void degree_kernel(const int* __restrict__ src, float* __restrict__ deg, int E) {
  int e = blockIdx.x * blockDim.x + threadIdx.x;
  if (e < E) unsafeAtomicAdd(&deg[src[e]], 1.0f);
}

// ---------------- per-edge coefficient ----------------
__global__ void coef_kernel(const int* __restrict__ src, const int* __restrict__ dst,
                            const float* __restrict__ w, const float* __restrict__ deg,
                            float* __restrict__ coef, int E) {
  int e = blockIdx.x * blockDim.x + threadIdx.x;
  if (e < E) {
    float ds = fmaxf(deg[src[e]], 1.0f);
    float dd = fmaxf(deg[dst[e]], 1.0f);
    coef[e] = rsqrtf(ds * dd) * w[e];
  }
}

// ---------------- gather * coef, scatter-add into agg ----------------
// 128 consecutive lanes handle one edge -> coalesced 512B gather + coalesced atomics.
__global__ void scatter_kernel(const float* __restrict__ x, const int* __restrict__ src,
                               const int* __restrict__ dst, const float* __restrict__ coef,
                               float* __restrict__ agg) {
  long i = (long)blockIdx.x * blockDim.x + threadIdx.x;
  int e = (int)(i >> 7);       // / 128
  int d = (int)(i & (DD - 1)); // % 128
  if (e < NE) {
    float c = coef[e];
    float v = x[(long)src[e] * DD + d] * c;
    unsafeAtomicAdd(&agg[(long)dst[e] * DD + d], v);
  }
}

// ---------------- FP32 WMMA GEMM: Out[M x 128] = act(A[M x 128] @ W[128 x 128] + bias) ----
// One wave computes one 16x16 output tile via 32 chained V_WMMA_F32_16X16X4_F32.
// A 16x4 layout : lanes 0-15 hold K = k0+0,k0+1 ; lanes 16-31 hold K = k0+2,k0+3 (ISA 7.12.2)
// B 4x16 layout : lane half selects K-row pair, lane&15 selects column
// C/D 16x16 f32 : VGPR i -> row i (lanes 0-15) / row i+8 (lanes 16-31), col = lane&15
template <bool RELU>
__global__ void gemm_wmma(const float* __restrict__ A, const float* __restrict__ W,
                          const float* __restrict__ bias, float* __restrict__ Out) {
  const int lane = threadIdx.x & 31;
  const int wave = threadIdx.x >> 5;
  const int tile = blockIdx.x * (blockDim.x >> 5) + wave;  // wave-uniform
  const int tm = tile >> 3;   // M tile (0..3124)
  const int tn = tile & 7;    // N tile (0..7)

  const int m15   = lane & 15;
  const int hi    = lane >> 4;   // 0 or 1 (half-wave)
  const int khalf = hi * 2;      // K sub-offset 0 or 2

  const int row = tm * 16 + m15;             // A row for this lane
  const int col = tn * 16 + m15;             // B/out column for this lane

  const float* arow = A + (long)row * DD + khalf;
  const float* bcol = W + (long)khalf * DD + col;

  v8f acc = {};
#pragma unroll
  for (int k0 = 0; k0 < DD; k0 += 4) {
    v2f a = *(const v2f*)(arow + k0);        // contiguous pair -> global_load_b64
    v2f b;
    b[0] = bcol[(long)k0 * DD];              // W[k0+khalf  ][col]
    b[1] = bcol[(long)k0 * DD + DD];         // W[k0+khalf+1][col]
    acc = __builtin_amdgcn_wmma_f32_16x16x4_f32(
        /*neg_a=*/false, a, /*neg_b=*/false, b,
        /*c_mod=*/(short)0, acc, /*reuse_a=*/false, /*reuse_b=*/false);
  }

  const float bv = bias[col];
  const int rbase = tm * 16 + hi * 8;        // rows rbase..rbase+7 for this half-wave
#pragma unroll
  for (int i = 0; i < 8; ++i) {
    float v = acc[i] + bv;
    if (RELU) v = fmaxf(v, 0.0f);
    Out[(long)(rbase + i) * DD + col] = v;
  }
}

extern "C" void kernel_launch(void* const* d_in, const int* in_sizes, int n_in,
                              void* d_out, int out_size, void* d_ws, size_t ws_size,
                              hipStream_t stream) {
  const float* x   = (const float*)d_in[0];
  const int*   src = (const int*)d_in[1];
  const int*   dst = (const int*)d_in[2];
  const float* ew  = (const float*)d_in[3];
  const float* W1  = (const float*)d_in[4];
  const float* b1  = (const float*)d_in[5];
  const float* W2  = (const float*)d_in[6];
  const float* b2  = (const float*)d_in[7];
  float* out = (float*)d_out;

  // workspace layout (bytes):
  //   deg  @ 0         : 50000 f32 (200 KB)
  //   coef @ 256 KB    : 800000 f32 (3.2 MB)
  //   agg  @ 4.25 MB   : 50000*128 f32 (25.6 MB)
  //   h    @ 30.25 MB  : 50000*128 f32 (25.6 MB)
  char* ws = (char*)d_ws;
  float* deg  = (float*)(ws);
  float* coef = (float*)(ws + (size_t)256 * 1024);
  float* agg  = (float*)(ws + (size_t)256 * 1024 + (size_t)4 * 1024 * 1024);
  float* h    = (float*)(ws + (size_t)256 * 1024 + (size_t)30 * 1024 * 1024);

  // 1) zero accumulators (must happen every call: graph replay reuses ws)
  zero_f32<<<(NN + 255) / 256, 256, 0, stream>>>(deg, NN);
  zero_f32<<<(NN * DD + 255) / 256, 256, 0, stream>>>(agg, NN * DD);

  // 2) degrees
  degree_kernel<<<(NE + 255) / 256, 256, 0, stream>>>(src, deg, NE);

  // 3) per-edge coefficients
  coef_kernel<<<(NE + 255) / 256, 256, 0, stream>>>(src, dst, ew, deg, coef, NE);

  // 4) gather-scale-scatter (bandwidth-bound phase: ~820 MB of traffic)
  scatter_kernel<<<(NE * DD) / 256, 256, 0, stream>>>(x, src, dst, coef, agg);

  // 5) FFN layer 1: h = relu(agg @ W1 + b1)   [3125*8 = 25000 wave-tiles, 4 waves/block]
  const int tiles = (NN / 16) * (DD / 16);
  gemm_wmma<true ><<<tiles / 4, 128, 0, stream>>>(agg, W1, b1, h);

  // 6) FFN layer 2: out = h @ W2 + b2
  gemm_wmma<false><<<tiles / 4, 128, 0, stream>>>(h, W2, b2, out);
}